// SumLayer_25262997635306
// MI455X (gfx1250) — compile-verified
//
#include <hip/hip_runtime.h>
#include <math.h>

#define N_NODES 4096
#define N_CHILD 16384
#define DEGREE  64
#define NNZ     (N_NODES * DEGREE)
#define BATCH   128
#define PSTR    288   // floats per edge-pair row: 2*128 + 32 pad -> pair p, p+1 use disjoint 32-bank windows

typedef __attribute__((ext_vector_type(2))) float v2f;
typedef __attribute__((ext_vector_type(8))) float v8f;

// ---------------------------------------------------------------------------
// Kernel A: XT[c*BATCH + b] = exp(child[b*N_CHILD + c])   (coalesced both ways)
// ---------------------------------------------------------------------------
__global__ __launch_bounds__(256) void k_exp_transpose(const float* __restrict__ child,
                                                       float* __restrict__ XT) {
    __shared__ float tile[32][33];
    const int tx = threadIdx.x;           // 0..31
    const int ty = threadIdx.y;           // 0..7
    const int c0 = blockIdx.x * 32;
    const int b0 = blockIdx.y * 32;
#pragma unroll
    for (int j = 0; j < 4; ++j) {
        int b = b0 + ty + j * 8;
        int c = c0 + tx;
        tile[ty + j * 8][tx] = expf(child[(size_t)b * N_CHILD + c]);
    }
    __syncthreads();
#pragma unroll
    for (int j = 0; j < 4; ++j) {
        int c = c0 + ty + j * 8;
        int b = b0 + tx;
        XT[(size_t)c * BATCH + b] = tile[tx][ty + j * 8];
    }
}

// ---------------------------------------------------------------------------
// Kernel B: per node n: mw = max_e lw, wexp[e] = exp(lw-mw), logZs[n]=log(sum)
// One wave32 per node; xor-shuffle reductions.
// ---------------------------------------------------------------------------
__global__ __launch_bounds__(32) void k_weights(const float* __restrict__ log_w,
                                                float* __restrict__ wexp,
                                                float* __restrict__ logZs) {
    const int n = blockIdx.x;
    const int l = threadIdx.x;            // 0..31
    const float* lw = log_w + (size_t)n * DEGREE;
    float a0 = lw[l];
    float a1 = lw[l + 32];
    float m = fmaxf(a0, a1);
#pragma unroll
    for (int off = 16; off >= 1; off >>= 1)
        m = fmaxf(m, __shfl_xor(m, off, 32));
    float e0 = expf(a0 - m);
    float e1 = expf(a1 - m);
    wexp[(size_t)n * DEGREE + l]      = e0;
    wexp[(size_t)n * DEGREE + 32 + l] = e1;
    float s = e0 + e1;
#pragma unroll
    for (int off = 16; off >= 1; off >>= 1)
        s += __shfl_xor(s, off, 32);
    if (l == 0) logZs[n] = logf(s);
}

// ---------------------------------------------------------------------------
// Kernel C: one block (128 thr = 4 waves) per node.
//  1. stage gathered 64x128 f32 tile of XT rows into LDS, edge-PAIR interleaved:
//       element (e, b) at Xs2[(e>>1)*PSTR + 2*b + (e&1)]
//     so each WMMA K-pair operand is one aligned ds_load_b64 (mov-free).
//  2. per wave: S[b] = sum_e wexp[e]*X[e,b] via chained V_WMMA_F32_16X16X4_F32
//     (A = weight 4-chunk broadcast over 16 rows; exact f32 MACs)
//  3. out[b, n] = log(S) - logZs[n]
// ---------------------------------------------------------------------------
__global__ __launch_bounds__(128) void k_sumnode(const float* __restrict__ XT,
                                                 const float* __restrict__ wexp,
                                                 const float* __restrict__ logZs,
                                                 const int* __restrict__ cols,
                                                 float* __restrict__ out) {
    __shared__ __align__(16) float Xs2[(DEGREE / 2) * PSTR];
    __shared__ __align__(16) float wls[DEGREE];
    __shared__ int icol[DEGREE];

    const int n = blockIdx.x;
    const int t = threadIdx.x;            // 0..127

    if (t < DEGREE) {
        icol[t] = cols[(size_t)n * DEGREE + t];
        wls[t]  = wexp[(size_t)n * DEGREE + t];
    }
    __syncthreads();

    // Gather + interleave: 1024 jobs; job f -> pair p = f>>5 (edges 2p,2p+1),
    // float4 column group q = f&31 (batch cols 4q..4q+3). Global reads stay
    // fully coalesced (lanes of a wave sweep consecutive float4 of one row).
#pragma unroll
    for (int it = 0; it < 8; ++it) {
        int f = it * 128 + t;             // 0..1023
        int p = f >> 5;                   // 0..31
        int q = f & 31;                   // 0..31
        float4 a = ((const float4*)(XT + (size_t)icol[2 * p]     * BATCH))[q]; // X[2p,   c..c+3]
        float4 b = ((const float4*)(XT + (size_t)icol[2 * p + 1] * BATCH))[q]; // X[2p+1, c..c+3]
        float* dst = &Xs2[p * PSTR + 8 * q];
        *(float4*)(dst)     = make_float4(a.x, b.x, a.y, b.y);
        *(float4*)(dst + 4) = make_float4(a.z, b.z, a.w, b.w);
    }
    __syncthreads();

    const int lane = t & 31;
    const int wv   = t >> 5;              // wave id 0..3 -> batch chunk of 32
    const int h    = lane >> 4;           // lane half supplies K = {2h, 2h+1} of each 4-chunk
    const int c16  = lane & 15;           // N (batch) column within 16-chunk

    float res[2];
#pragma unroll
    for (int half = 0; half < 2; ++half) {
        const int b0 = wv * 32 + half * 16;
        v8f acc = {0.f, 0.f, 0.f, 0.f, 0.f, 0.f, 0.f, 0.f};
#pragma unroll
        for (int i = 0; i < 16; ++i) {
            // A pair: w[4i+2h], w[4i+2h+1]  (contiguous, 8B aligned)
            v2f A = *(const v2f*)&wls[4 * i + 2 * h];
            // B pair: X[4i+2h, b0+c16], X[4i+2h+1, b0+c16] -> pair index 2i+h
            v2f B = *(const v2f*)&Xs2[(2 * i + h) * PSTR + 2 * (b0 + c16)];
            acc = __builtin_amdgcn_wmma_f32_16x16x4_f32(
                /*neg_a=*/false, A, /*neg_b=*/false, B,
                /*c_mod=*/(short)0, acc, /*reuse_a=*/false, /*reuse_b=*/false);
        }
        res[half] = acc[0];               // all 16 D rows identical; holds S[b0+c16]
    }
    // lanes 0-15 carry half 0 (N=lane), lanes 16-31 half 1 (N=lane-16)
    const float S = (h == 0) ? res[0] : res[1];
    const int b = wv * 32 + lane;
    out[(size_t)b * N_NODES + n] = logf(S) - logZs[n];
}

// ---------------------------------------------------------------------------
extern "C" void kernel_launch(void* const* d_in, const int* in_sizes, int n_in,
                              void* d_out, int out_size, void* d_ws, size_t ws_size,
                              hipStream_t stream) {
    const float* child = (const float*)d_in[0];   // [BATCH, N_CHILD]
    const float* log_w = (const float*)d_in[1];   // [NNZ]
    // d_in[2] = rows: structure is implicit (n = e / DEGREE), unused
    const int*   cols  = (const int*)d_in[3];     // [NNZ]
    float* out = (float*)d_out;                   // [BATCH, N_NODES]

    float* XT    = (float*)d_ws;                  // [N_CHILD, BATCH]  8 MB
    float* wexp  = XT + (size_t)N_CHILD * BATCH;  // [NNZ]             1 MB
    float* logZs = wexp + NNZ;                    // [N_NODES]         16 KB

    k_exp_transpose<<<dim3(N_CHILD / 32, BATCH / 32), dim3(32, 8), 0, stream>>>(child, XT);
    k_weights<<<N_NODES, 32, 0, stream>>>(log_w, wexp, logZs);
    k_sumnode<<<N_NODES, 128, 0, stream>>>(XT, wexp, logZs, cols, out);
}